// StochasticDropout_70832600646053
// MI455X (gfx1250) — compile-verified
//
#include <hip/hip_runtime.h>

// StochasticDropout: out = x * exp(-sigma^2/2 + sigma*eps), sigma = 0.5
// Pure streaming elementwise op: 805 MB traffic / 23.3 TB/s => ~34.6 us floor.
// Strategy: b128 NT loads/stores (working set >> 192MB L2, zero reuse),
// v_exp_f32 via exp2 with log2(e) folded into one FMA, wave32 vec4 threads.

typedef float v4f __attribute__((ext_vector_type(4)));

// exp(a) = exp2(a * log2e); fold constants:
//   S = sigma * log2e, A = -(sigma^2/2) * log2e
#define SDO_S 0.72134752044448170367996234050095f   // 0.5 * log2(e)
#define SDO_A (-0.18033688011112042591999058512524f) // -0.125 * log2(e)

__device__ __forceinline__ float sdo_one(float x, float e) {
    // v_fma_f32 + v_exp_f32 + v_mul_f32
    return x * __builtin_amdgcn_exp2f(__builtin_fmaf(e, SDO_S, SDO_A));
}

__global__ __launch_bounds__(256) void StochasticDropout_vec4(
    const v4f* __restrict__ x, const v4f* __restrict__ eps,
    v4f* __restrict__ out, long long n4) {
    long long i = (long long)blockIdx.x * blockDim.x + threadIdx.x;
    if (i >= n4) return;

    // 128-bit non-temporal loads: global_load_b128 th:NT
    v4f xv = __builtin_nontemporal_load(&x[i]);
    v4f ev = __builtin_nontemporal_load(&eps[i]);

    v4f o;
    o.x = sdo_one(xv.x, ev.x);
    o.y = sdo_one(xv.y, ev.y);
    o.z = sdo_one(xv.z, ev.z);
    o.w = sdo_one(xv.w, ev.w);

    // 128-bit non-temporal store: global_store_b128 th:NT
    __builtin_nontemporal_store(o, &out[i]);
}

__global__ __launch_bounds__(256) void StochasticDropout_tail(
    const float* __restrict__ x, const float* __restrict__ eps,
    float* __restrict__ out, long long n) {
    long long i = (long long)blockIdx.x * blockDim.x + threadIdx.x;
    if (i >= n) return;
    float xv = __builtin_nontemporal_load(&x[i]);
    float ev = __builtin_nontemporal_load(&eps[i]);
    __builtin_nontemporal_store(sdo_one(xv, ev), &out[i]);
}

extern "C" void kernel_launch(void* const* d_in, const int* in_sizes, int n_in,
                              void* d_out, int out_size, void* d_ws, size_t ws_size,
                              hipStream_t stream) {
    const float* x   = (const float*)d_in[0];
    const float* eps = (const float*)d_in[1];
    float* out       = (float*)d_out;

    const long long n  = (long long)in_sizes[0];
    const long long n4 = n >> 2;          // vec4 chunks (n = 2^26 here, so exact)
    const long long rem = n - (n4 << 2);

    if (n4 > 0) {
        const long long blocks = (n4 + 255) / 256;
        StochasticDropout_vec4<<<dim3((unsigned int)blocks), dim3(256), 0, stream>>>(
            (const v4f*)x, (const v4f*)eps, (v4f*)out, n4);
    }
    if (rem > 0) {
        const long long off = n4 << 2;
        StochasticDropout_tail<<<dim3(1), dim3(256), 0, stream>>>(
            x + off, eps + off, out + off, rem);
    }
}